// MfccDecoder_30906584662742
// MI455X (gfx1250) — compile-verified
//
#include <hip/hip_runtime.h>

// ---------------------------------------------------------------------------
// CDNA5 (gfx1250) fp32 WMMA helpers: V_WMMA_F32_16X16X4_F32
//   A: 16x4 f32  -> v2f per lane (VGPR0=K{0,2}, VGPR1=K{1,3}; lanes16-31 = K+2)
//   B: 4x16 f32  -> v2f per lane (mirror of A; N = lane&15)
//   C/D: 16x16   -> v8f per lane (VGPR v: M=v (lanes0-15) / M=v+8 (lanes16-31), N=lane&15)
// ---------------------------------------------------------------------------
typedef __attribute__((ext_vector_type(2))) float v2f;
typedef __attribute__((ext_vector_type(8))) float v8f;

static __device__ __forceinline__ v8f wmma_f32x4(v2f a, v2f b, v8f c) {
  return __builtin_amdgcn_wmma_f32_16x16x4_f32(
      /*neg_a=*/false, a, /*neg_b=*/false, b,
      /*c_mod=*/(short)0, c, /*reuse_a=*/false, /*reuse_b=*/false);
}

static __device__ __forceinline__ float sigmoidf_(float x) {
  return 1.0f / (1.0f + __expf(-x));
}
static __device__ __forceinline__ float scalef_(float x) {
  // 2 * sigmoid(x)^ln(10) + 1e-7
  float s = sigmoidf_(x);
  return 2.0f * __powf(s, 2.302585093f) + 1e-7f;
}
static __device__ __forceinline__ float leakyf_(float x) {
  return x > 0.0f ? x : 0.01f * x;
}

#define H512 512
#define TBT  3200      // B*T
#define NTOK 102400    // B*T*32
#define LSIG 64000     // T*BLOCK
#define PI_F 3.14159265358979f
#define KCH  128       // GEMM K-chunk staged in LDS

// ---------------------------------------------------------------------------
// WMMA GEMM: C[M,N] = A[M,K] @ W[N,K]^T + bias[N]
// 256-thread block (8 waves): one 16-row M-tile, 128 N-columns (16 per wave).
// A staged through LDS in 16xKCH chunks; A frags via ds_load, B via global b64.
// Full chunks use a compile-time-bound, unrolled WMMA loop so the compiler
// software-pipelines loads (staggered s_wait_loadcnt) across the WMMA stream.
// ---------------------------------------------------------------------------
__global__ __launch_bounds__(256) void k_gemm(
    const float* __restrict__ A, const float* __restrict__ W,
    const float* __restrict__ bias, float* __restrict__ C,
    int M, int N, int K)
{
  __shared__ float As[16 * KCH];            // 8 KB
  const int mt   = blockIdx.x * 16;
  const int nt0  = blockIdx.y * 128;
  const int tid  = threadIdx.x;
  const int lane = tid & 31;
  const int wv   = tid >> 5;                 // wave 0..7
  const int r    = lane & 15;
  const int kh   = (lane >> 4) * 2;          // K sub-offset: 0 or 2

  const int ntile = nt0 + wv * 16;
  int nrow = ntile + r; if (nrow >= N) nrow = N - 1;   // clamp (no EXEC divergence)
  const float* wrow = W + (size_t)nrow * K;

  // cooperative A stage: thread loads 8 floats of row (tid>>4), cols (tid&15)*8
  int arow_i = mt + (tid >> 4); if (arow_i >= M) arow_i = M - 1;
  const float* arow_g = A + (size_t)arow_i * K + (tid & 15) * 8;
  float* as_st = As + (tid >> 4) * KCH + (tid & 15) * 8;

  v8f acc = {};
  for (int kc = 0; kc < K; kc += KCH) {
    int kl = K - kc; if (kl > KCH) kl = KCH;          // uniform remainder
    const bool ldok = ((tid & 15) * 8) < kl;
    float4 a0{}, a1{};
    if (ldok) {
      a0 = *(const float4*)(arow_g + kc);
      a1 = *(const float4*)(arow_g + kc + 4);
    }
    if (kc + KCH < K)                                  // prefetch next B chunk
      __builtin_prefetch((const void*)(wrow + kc + KCH), 0, 1);
    __syncthreads();                                   // prior chunk fully read
    if (ldok) {
      *(float4*)as_st       = a0;
      *(float4*)(as_st + 4) = a1;
    }
    __syncthreads();
    const float* wp = wrow + kc + kh;
    const float* ap = As + r * KCH + kh;
    if (kl == KCH) {                                   // uniform: main path
#pragma unroll 8
      for (int k0 = 0; k0 < KCH; k0 += 4) {
        v2f a; a.x = ap[k0]; a.y = ap[k0 + 1];
        float2 bv = *(const float2*)(wp + k0);
        v2f b; b.x = bv.x; b.y = bv.y;
        acc = wmma_f32x4(a, b, acc);
      }
    } else {                                           // K remainder (rare)
      for (int k0 = 0; k0 < kl; k0 += 4) {
        v2f a; a.x = ap[k0]; a.y = ap[k0 + 1];
        float2 bv = *(const float2*)(wp + k0);
        v2f b; b.x = bv.x; b.y = bv.y;
        acc = wmma_f32x4(a, b, acc);
      }
    }
  }

  const int n  = ntile + (lane & 15);
  const int mb = mt + (lane >> 4) * 8;
  if (n < N) {
    const float bv = bias[n];
#pragma unroll
    for (int v = 0; v < 8; ++v) {
      int m = mb + v;
      if (m < M) C[(size_t)m * N + n] = acc[v] + bv;
    }
  }
}

// ---------------------------------------------------------------------------
// LayerNorm(512) + LeakyReLU(0.01), in place, one block per row
// ---------------------------------------------------------------------------
__global__ __launch_bounds__(256) void k_ln_leaky(
    float* __restrict__ X, const float* __restrict__ g, const float* __restrict__ e)
{
  __shared__ float s1[256], s2[256];
  float* xr = X + (size_t)blockIdx.x * H512;
  const int tid = threadIdx.x;
  float v0 = xr[tid], v1 = xr[tid + 256];
  s1[tid] = v0 + v1;
  s2[tid] = v0 * v0 + v1 * v1;
  __syncthreads();
  for (int off = 128; off > 0; off >>= 1) {
    if (tid < off) { s1[tid] += s1[tid + off]; s2[tid] += s2[tid + off]; }
    __syncthreads();
  }
  const float mean = s1[0] * (1.0f / 512.0f);
  const float var  = s2[0] * (1.0f / 512.0f) - mean * mean;
  const float inv  = rsqrtf(var + 1e-5f);
  float y0 = (v0 - mean) * inv * g[tid] + e[tid];
  float y1 = (v1 - mean) * inv * g[tid + 256] + e[tid + 256];
  xr[tid]       = leakyf_(y0);
  xr[tid + 256] = leakyf_(y1);
}

// ---------------------------------------------------------------------------
// Input MLP layer 0: X0[tok] = leaky(LN(feat_scalar * w0 + b0))  (512-wide)
// ---------------------------------------------------------------------------
__global__ __launch_bounds__(256) void k_feat0(
    const float* __restrict__ pitch, const float* __restrict__ loud,
    const float* __restrict__ mfccs, const float* __restrict__ w0,
    const float* __restrict__ b0, const float* __restrict__ g0,
    const float* __restrict__ e0, float* __restrict__ X)
{
  __shared__ float s1[256], s2[256];
  const int it = blockIdx.x;     // token = bt*32 + c
  const int c  = it & 31;
  const int bt = it >> 5;
  float f;
  if (c == 0)      f = pitch[bt];
  else if (c == 1) f = loud[bt];
  else             f = mfccs[bt * 32 + c];   // feats[c] = mfccs[...,c], c>=2

  const int tid = threadIdx.x;
  float v0 = f * w0[tid]       + b0[tid];
  float v1 = f * w0[tid + 256] + b0[tid + 256];
  s1[tid] = v0 + v1;
  s2[tid] = v0 * v0 + v1 * v1;
  __syncthreads();
  for (int off = 128; off > 0; off >>= 1) {
    if (tid < off) { s1[tid] += s1[tid + off]; s2[tid] += s2[tid + off]; }
    __syncthreads();
  }
  const float mean = s1[0] * (1.0f / 512.0f);
  const float var  = s2[0] * (1.0f / 512.0f) - mean * mean;
  const float inv  = rsqrtf(var + 1e-5f);
  float y0 = (v0 - mean) * inv * g0[tid] + e0[tid];
  float y1 = (v1 - mean) * inv * g0[tid + 256] + e0[tid + 256];
  float* xr = X + (size_t)it * H512;
  xr[tid]       = leakyf_(y0);
  xr[tid + 256] = leakyf_(y1);
}

// ---------------------------------------------------------------------------
// GRU recurrence: single 1024-thread workgroup (32 waves), h in LDS,
// gh = h @ whh^T via WMMA (A operands from LDS), gates fused.
// ---------------------------------------------------------------------------
__global__ __launch_bounds__(1024) void k_gru(
    const float* __restrict__ XI, const float* __restrict__ whh,
    const float* __restrict__ bhh, float* __restrict__ G)
{
  __shared__ float hs[8 * 512];      // 16 KB
  __shared__ float ghs[8 * 1536];    // 48 KB
  const int tid  = threadIdx.x;
  const int lane = tid & 31;
  const int w    = tid >> 5;         // wave 0..31
  for (int i = tid; i < 8 * 512; i += 1024) hs[i] = 0.0f;
  __syncthreads();

  const int r    = lane & 15;
  const int kh   = (lane >> 4) * 2;
  const int hrow = r & 7;            // clamp A rows (M=8 real rows)

  for (int t = 0; t < 400; ++t) {
    // --- gh = h @ whh^T + bhh (96 N-tiles over 32 waves) ---
    for (int tt = 0; tt < 3; ++tt) {
      const int ntile = (w + tt * 32) * 16;
      const float* wrow = whh + (size_t)(ntile + r) * 512 + kh;
      const float* ap   = hs + hrow * 512 + kh;
      v8f acc = {};
#pragma unroll 8
      for (int k0 = 0; k0 < 512; k0 += 4) {
        v2f a; a.x = ap[k0]; a.y = ap[k0 + 1];
        float2 bv = *(const float2*)(wrow + k0);
        v2f b; b.x = bv.x; b.y = bv.y;
        acc = wmma_f32x4(a, b, acc);
      }
      const int nc = ntile + (lane & 15);
      const int mb = (lane >> 4) * 8;
      const float bb = bhh[nc];
#pragma unroll
      for (int v = 0; v < 8; ++v) {
        int m = mb + v;
        if (m < 8) ghs[m * 1536 + nc] = acc[v] + bb;
      }
    }
    __syncthreads();
    // --- gates ---
    for (int i = tid; i < 8 * 512; i += 1024) {
      const int b = i >> 9, j = i & 511;
      const size_t row = (size_t)(b * 400 + t);
      const float ir = XI[row * 1536 + j];
      const float iz = XI[row * 1536 + 512 + j];
      const float in = XI[row * 1536 + 1024 + j];
      const float hr = ghs[b * 1536 + j];
      const float hz = ghs[b * 1536 + 512 + j];
      const float hn = ghs[b * 1536 + 1024 + j];
      const float rr = sigmoidf_(ir + hr);
      const float zz = sigmoidf_(iz + hz);
      const float nn = tanhf(in + rr * hn);
      const float h  = (1.0f - zz) * nn + zz * hs[i];
      hs[i] = h;
      G[row * 512 + j] = h;
    }
    __syncthreads();
  }
}

// ---------------------------------------------------------------------------
// Pack out-MLP input: [g(512) | pitch | loudness | mfccs[:30]] -> 544
// ---------------------------------------------------------------------------
__global__ void k_pack(
    const float* __restrict__ G, const float* __restrict__ pitch,
    const float* __restrict__ loud, const float* __restrict__ mfccs,
    float* __restrict__ H2)
{
  int i = blockIdx.x * blockDim.x + threadIdx.x;
  if (i >= TBT * 544) return;
  int row = i / 544, c = i % 544;
  float v;
  if (c < 512)       v = G[row * 512 + c];
  else if (c == 512) v = pitch[row];
  else if (c == 513) v = loud[row];
  else               v = mfccs[row * 32 + (c - 514)];
  H2[i] = v;
}

// ---------------------------------------------------------------------------
// Harmonic amplitudes: scale + nyquist mask + normalize, per frame
// ---------------------------------------------------------------------------
__global__ __launch_bounds__(128) void k_amps(
    const float* __restrict__ P0, const float* __restrict__ pitch,
    float* __restrict__ AN)
{
  __shared__ float s[128];
  const int row = blockIdx.x;
  const int k   = threadIdx.x;
  const float p = pitch[row];
  float a = 0.0f;
  if (k < 100) {
    float sc = scalef_(P0[row * 101 + 1 + k]);
    float aa = ((p * (float)(k + 1)) < 8000.0f ? 1.0f : 0.0f) + 1e-4f;
    a = sc * aa;
  }
  s[k] = a;
  __syncthreads();
  for (int off = 64; off > 0; off >>= 1) {
    if (k < off) s[k] += s[k + off];
    __syncthreads();
  }
  const float total = scalef_(P0[row * 101]);
  if (k < 100) AN[row * 100 + k] = a / s[0] * total;
}

// ---------------------------------------------------------------------------
// Phase prefix per batch: cum_excl[t] = sum_{u<t} 160*w_u,  w = 2*pi*pitch/SR
// ---------------------------------------------------------------------------
__global__ void k_phase(const float* __restrict__ pitch,
                        float* __restrict__ CUM, float* __restrict__ WF)
{
  const int b = threadIdx.x;
  if (b >= 8) return;
  float acc = 0.0f;
  for (int t = 0; t < 400; ++t) {
    float w = 2.0f * PI_F * pitch[b * 400 + t] * (1.0f / 16000.0f);
    CUM[b * 400 + t] = acc;
    WF[b * 400 + t]  = w;
    acc += 160.0f * w;
  }
}

// ---------------------------------------------------------------------------
// Harmonic synthesis: harm[b,s] = sum_k sin(omega_s * k) * amps[b,t,k-1]
// ---------------------------------------------------------------------------
__global__ __launch_bounds__(256) void k_harm(
    const float* __restrict__ CUM, const float* __restrict__ WF,
    const float* __restrict__ AN, float* __restrict__ HARM)
{
  int gid = blockIdx.x * blockDim.x + threadIdx.x;
  if (gid >= 8 * LSIG) return;
  const int b = gid / LSIG;
  const int s = gid % LSIG;
  const int t = s / 160;
  const int i = s % 160;
  const float om = CUM[b * 400 + t] + (float)(i + 1) * WF[b * 400 + t];
  const float* am = AN + (size_t)(b * 400 + t) * 100;
  float acc = 0.0f;
#pragma unroll 4
  for (int k = 1; k <= 100; ++k)
    acc += __sinf(om * (float)k) * am[k - 1];
  HARM[gid] = acc;
}

// ---------------------------------------------------------------------------
// amp_to_ir: irfft(65 bins -> 128) . roll . hann . pad(160) . roll, per frame
// ---------------------------------------------------------------------------
__global__ __launch_bounds__(160) void k_ir(
    const float* __restrict__ P1, float* __restrict__ IMPF)
{
  __shared__ float af[65];
  const int row = blockIdx.x;
  const int n   = threadIdx.x;
  if (n < 65) af[n] = scalef_(P1[row * 65 + n] - 5.0f);
  __syncthreads();
  const int j = (n + 64) % 160;
  float val = 0.0f;
  if (j < 128) {
    const int m = (j + 64) & 127;
    float acc = af[0];
    for (int f = 1; f < 64; ++f)
      acc += 2.0f * af[f] * __cosf(2.0f * PI_F * (float)(f * m) * (1.0f / 128.0f));
    acc += af[64] * ((m & 1) ? -1.0f : 1.0f);
    const float ir0 = acc * (1.0f / 128.0f);
    const float win = 0.5f * (1.0f - __cosf(2.0f * PI_F * (float)j * (1.0f / 128.0f)));
    val = ir0 * win;
  }
  IMPF[row * 160 + n] = val;
}

// ---------------------------------------------------------------------------
// Per-frame causal conv of (noise*2-1) with impulse, add harmonic -> sig
// ---------------------------------------------------------------------------
__global__ __launch_bounds__(160) void k_noise(
    const float* __restrict__ noise, const float* __restrict__ IMPF,
    const float* __restrict__ HARM, float* __restrict__ SIG)
{
  __shared__ float xs[160], ks[160];
  const int row = blockIdx.x;
  const int q   = threadIdx.x;
  xs[q] = noise[row * 160 + q] * 2.0f - 1.0f;
  ks[q] = IMPF[row * 160 + q];
  __syncthreads();
  float acc = 0.0f;
  for (int p = 0; p <= q; ++p) acc += ks[p] * xs[q - p];
  const int b = row / 400, t = row % 400;
  const int s = b * LSIG + t * 160 + q;
  SIG[s] = HARM[s] + acc;
}

// ---------------------------------------------------------------------------
// Reverb impulse: rev_noise * exp(-softplus(-dec)*t*500) * sigmoid(wet); [0]=1
// ---------------------------------------------------------------------------
__global__ void k_revimp(const float* __restrict__ rn,
                         const float* __restrict__ dec, const float* __restrict__ wet,
                         float* __restrict__ IREV)
{
  int i = blockIdx.x * blockDim.x + threadIdx.x;
  if (i >= 16000) return;
  const float sp = log1pf(__expf(-dec[0]));
  const float wt = sigmoidf_(wet[0]);
  float v = rn[i] * __expf(-sp * ((float)i * (1.0f / 16000.0f)) * 500.0f) * wt;
  if (i == 0) v = 1.0f;
  IREV[i] = v;
}

// ---------------------------------------------------------------------------
// Reverb: out[b,q] = sum_{p<=min(q,15999)} imp[p] * sig[b,q-p], LDS-tiled taps
// ---------------------------------------------------------------------------
__global__ __launch_bounds__(256) void k_reverb(
    const float* __restrict__ SIG, const float* __restrict__ IREV,
    float* __restrict__ OUT)
{
  __shared__ float kb[256];
  const int b   = blockIdx.x / 250;
  const int q0  = (blockIdx.x % 250) * 256;
  const int tid = threadIdx.x;
  const int q   = q0 + tid;
  const float* sb = SIG + (size_t)b * LSIG;
  float acc = 0.0f;
  for (int p0 = 0; p0 < 16000; p0 += 256) {
    if (p0 > q0 + 255) break;                 // uniform early-out
    kb[tid] = (p0 + tid < 16000) ? IREV[p0 + tid] : 0.0f;
    __syncthreads();
#pragma unroll 4
    for (int pp = 0; pp < 256; ++pp) {
      const int idx = q - (p0 + pp);
      if (idx >= 0) acc += kb[pp] * sb[idx];
    }
    __syncthreads();
  }
  OUT[(size_t)b * LSIG + q] = acc;
}

// ---------------------------------------------------------------------------
extern "C" void kernel_launch(void* const* d_in, const int* in_sizes, int n_in,
                              void* d_out, int out_size, void* d_ws, size_t ws_size,
                              hipStream_t stream)
{
  const float* pitch   = (const float*)d_in[0];
  const float* loud    = (const float*)d_in[1];
  const float* mfccs   = (const float*)d_in[2];
  const float* noise   = (const float*)d_in[3];
  const float* in_w0   = (const float*)d_in[4];
  const float* in_b0   = (const float*)d_in[5];
  const float* in_g0   = (const float*)d_in[6];
  const float* in_e0   = (const float*)d_in[7];
  const float* in_w12  = (const float*)d_in[8];
  const float* in_b12  = (const float*)d_in[9];
  const float* in_g12  = (const float*)d_in[10];
  const float* in_e12  = (const float*)d_in[11];
  const float* gru_wih = (const float*)d_in[12];
  const float* gru_whh = (const float*)d_in[13];
  const float* gru_bih = (const float*)d_in[14];
  const float* gru_bhh = (const float*)d_in[15];
  const float* out_w0  = (const float*)d_in[16];
  const float* out_b0  = (const float*)d_in[17];
  const float* out_g0  = (const float*)d_in[18];
  const float* out_e0  = (const float*)d_in[19];
  const float* out_w12 = (const float*)d_in[20];
  const float* out_b12 = (const float*)d_in[21];
  const float* out_g12 = (const float*)d_in[22];
  const float* out_e12 = (const float*)d_in[23];
  const float* proj0_w = (const float*)d_in[24];
  const float* proj0_b = (const float*)d_in[25];
  const float* proj1_w = (const float*)d_in[26];
  const float* proj1_b = (const float*)d_in[27];
  const float* rev_n   = (const float*)d_in[28];
  const float* rev_dec = (const float*)d_in[29];
  const float* rev_wet = (const float*)d_in[30];

  char* ws = (char*)d_ws;
  size_t off = 0;
  auto carve = [&](size_t bytes) -> float* {
    float* p = (float*)(ws + off);
    off += (bytes + 255) & ~(size_t)255;
    return p;
  };
  float* Xa   = carve((size_t)NTOK * 512 * 4);
  float* Xb   = carve((size_t)NTOK * 512 * 4);
  float* XI   = carve((size_t)TBT * 1536 * 4);
  float* Gm   = carve((size_t)TBT * 512 * 4);
  float* H2   = carve((size_t)TBT * 544 * 4);
  float* Ya   = carve((size_t)TBT * 512 * 4);
  float* Yb   = carve((size_t)TBT * 512 * 4);
  float* P0   = carve((size_t)TBT * 101 * 4);
  float* P1   = carve((size_t)TBT * 65 * 4);
  float* AN   = carve((size_t)TBT * 100 * 4);
  float* CUM  = carve((size_t)TBT * 4);
  float* WF   = carve((size_t)TBT * 4);
  float* HARM = carve((size_t)8 * LSIG * 4);
  float* SIG  = carve((size_t)8 * LSIG * 4);
  float* IMPF = carve((size_t)TBT * 160 * 4);
  float* IREV = carve((size_t)16000 * 4);

  const dim3 b256(256);

  // input MLP (tokens = 102400, H = 512)
  k_feat0<<<NTOK, 256, 0, stream>>>(pitch, loud, mfccs, in_w0, in_b0, in_g0, in_e0, Xa);
  k_gemm<<<dim3(NTOK / 16, 4), b256, 0, stream>>>(Xa, in_w12,             in_b12,       Xb, NTOK, 512, 512);
  k_ln_leaky<<<NTOK, 256, 0, stream>>>(Xb, in_g12,       in_e12);
  k_gemm<<<dim3(NTOK / 16, 4), b256, 0, stream>>>(Xb, in_w12 + 512 * 512, in_b12 + 512, Xa, NTOK, 512, 512);
  k_ln_leaky<<<NTOK, 256, 0, stream>>>(Xa, in_g12 + 512, in_e12 + 512);

  // GRU input projection (hidden = Xa viewed as [3200, 16384]) + recurrence
  k_gemm<<<dim3(TBT / 16, 12), b256, 0, stream>>>(Xa, gru_wih, gru_bih, XI, TBT, 1536, 16384);
  k_gru<<<1, 1024, 0, stream>>>(XI, gru_whh, gru_bhh, Gm);

  // output MLP
  k_pack<<<(TBT * 544 + 255) / 256, 256, 0, stream>>>(Gm, pitch, loud, mfccs, H2);
  k_gemm<<<dim3(TBT / 16, 4), b256, 0, stream>>>(H2, out_w0,              out_b0,        Ya, TBT, 512, 544);
  k_ln_leaky<<<TBT, 256, 0, stream>>>(Ya, out_g0,        out_e0);
  k_gemm<<<dim3(TBT / 16, 4), b256, 0, stream>>>(Ya, out_w12,             out_b12,       Yb, TBT, 512, 512);
  k_ln_leaky<<<TBT, 256, 0, stream>>>(Yb, out_g12,       out_e12);
  k_gemm<<<dim3(TBT / 16, 4), b256, 0, stream>>>(Yb, out_w12 + 512 * 512, out_b12 + 512, Ya, TBT, 512, 512);
  k_ln_leaky<<<TBT, 256, 0, stream>>>(Ya, out_g12 + 512, out_e12 + 512);

  // projections
  k_gemm<<<dim3(TBT / 16, 1), b256, 0, stream>>>(Ya, proj0_w, proj0_b, P0, TBT, 101, 512);
  k_gemm<<<dim3(TBT / 16, 1), b256, 0, stream>>>(Ya, proj1_w, proj1_b, P1, TBT, 65, 512);

  // DSP tail
  k_amps<<<TBT, 128, 0, stream>>>(P0, pitch, AN);
  k_phase<<<1, 32, 0, stream>>>(pitch, CUM, WF);
  k_harm<<<(8 * LSIG + 255) / 256, 256, 0, stream>>>(CUM, WF, AN, HARM);
  k_ir<<<TBT, 160, 0, stream>>>(P1, IMPF);
  k_noise<<<TBT, 160, 0, stream>>>(noise, IMPF, HARM, SIG);
  k_revimp<<<(16000 + 255) / 256, 256, 0, stream>>>(rev_n, rev_dec, rev_wet, IREV);
  k_reverb<<<8 * 250, 256, 0, stream>>>(SIG, IREV, (float*)d_out);
}